// LogSinkhorn_77481210020131
// MI455X (gfx1250) — compile-verified
//
#include <hip/hip_runtime.h>
#include <math.h>

#define BATCH 64
#define N 1024
#define NUM_ITERS 20
#define ROWC 4               // row chunks for the column pass
#define CHUNK (N / ROWC)     // 256 rows per chunk (== blockDim)

typedef __attribute__((ext_vector_type(2))) float v2f;
typedef __attribute__((ext_vector_type(8))) float v8f;

// ---------------------------------------------------------------------------
// u[b,i] = logsumexp_j( logits[b,i,j] - v[b,j] )
// One 256-thread block per row. Row max via wave shuffles; sum of the 1024
// exponentials via V_WMMA_F32_16X16X4_F32 with an all-ones A matrix:
// D = ones(16x4) x B(4x16) + C  => every row of D holds the column sums of B,
// so summing lanes 0..15 of D's first VGPR gives the total of all 64 fed
// values (invariant to the exact B lane striping). Two accumulated WMMAs
// consume the 128 exp values held by each wave.
// ---------------------------------------------------------------------------
__global__ __launch_bounds__(256) void rowLSE(const float* __restrict__ logits,
                                              const float* __restrict__ v,
                                              float* __restrict__ u) {
  const int row = blockIdx.x, b = blockIdx.y, tid = threadIdx.x;
  const float4* rp = (const float4*)(logits + ((size_t)b * N + row) * N);
  const float4* vp = (const float4*)(v + (size_t)b * N);
  float4 x  = rp[tid];
  float4 vv = vp[tid];
  x.x -= vv.x; x.y -= vv.y; x.z -= vv.z; x.w -= vv.w;

  __shared__ float sm[16];
  const int wave = tid >> 5;

  // ---- block max ----
  float m4 = fmaxf(fmaxf(x.x, x.y), fmaxf(x.z, x.w));
  #pragma unroll
  for (int msk = 16; msk >= 1; msk >>= 1)
    m4 = fmaxf(m4, __shfl_xor(m4, msk, 32));
  if ((tid & 31) == 0) sm[wave] = m4;
  __syncthreads();
  float M = sm[0];
  #pragma unroll
  for (int i = 1; i < 8; ++i) M = fmaxf(M, sm[i]);

  // ---- exponentials relative to row max ----
  float e0 = expf(x.x - M), e1 = expf(x.y - M);
  float e2 = expf(x.z - M), e3 = expf(x.w - M);

  // ---- per-wave sum of 128 exps on the matrix unit ----
  v2f a;   a[0] = 1.0f;  a[1] = 1.0f;        // all-ones 16x4 A matrix
  v2f b01; b01[0] = e0;  b01[1] = e1;
  v2f b23; b23[0] = e2;  b23[1] = e3;
  v8f acc = {0.f, 0.f, 0.f, 0.f, 0.f, 0.f, 0.f, 0.f};
  acc = __builtin_amdgcn_wmma_f32_16x16x4_f32(false, a, false, b01,
                                              (short)0, acc, false, false);
  acc = __builtin_amdgcn_wmma_f32_16x16x4_f32(false, a, false, b23,
                                              (short)0, acc, false, false);
  // lanes 0..15 (and 16..31 identically) hold 16 column sums in acc[0];
  // fold the 16-lane half to get the per-wave total.
  float wsum = acc[0];
  #pragma unroll
  for (int msk = 8; msk >= 1; msk >>= 1)
    wsum += __shfl_xor(wsum, msk, 32);
  if ((tid & 31) == 0) sm[8 + wave] = wsum;
  __syncthreads();

  if (tid == 0) {
    float S = 0.f;
    #pragma unroll
    for (int i = 0; i < 8; ++i) S += sm[8 + i];
    u[(size_t)b * N + row] = M + logf(S);
  }
}

// ---------------------------------------------------------------------------
// Partial column LSE over a 256-row chunk: thread t owns column j, streaming
// coalesced rows. Online one-exp logsumexp with 4 independent accumulator
// chains (i%4) so loads of later rows issue while exp chains retire.
// ---------------------------------------------------------------------------
__global__ __launch_bounds__(256) void colLSEpart(const float* __restrict__ logits,
                                                  const float* __restrict__ u,
                                                  float* __restrict__ pm,
                                                  float* __restrict__ ps) {
  const int tid = threadIdx.x;
  const int j   = blockIdx.x * 256 + tid;
  const int rc  = blockIdx.y;
  const int b   = blockIdx.z;

  __shared__ float su[CHUNK];
  su[tid] = u[(size_t)b * N + rc * CHUNK + tid];
  __syncthreads();

  const float* base = logits + ((size_t)b * N + (size_t)rc * CHUNK) * N + j;

  float m[4], s[4];
  #pragma unroll
  for (int k = 0; k < 4; ++k) { m[k] = -INFINITY; s[k] = 0.f; }

  for (int i = 0; i < CHUNK; i += 4) {
    if (i + 8 < CHUNK)
      __builtin_prefetch(base + (size_t)(i + 8) * N, 0, 1);
    #pragma unroll
    for (int k = 0; k < 4; ++k) {
      float x  = base[(size_t)(i + k) * N] - su[i + k];
      float hi = fmaxf(x, m[k]);
      float lo = fminf(x, m[k]);
      float d  = expf(lo - hi);                       // single exp per element
      s[k] = (x > m[k]) ? fmaf(s[k], d, 1.0f) : (s[k] + d);
      m[k] = hi;
    }
  }
  // merge the 4 chains
  float M = fmaxf(fmaxf(m[0], m[1]), fmaxf(m[2], m[3]));
  float S = 0.f;
  #pragma unroll
  for (int k = 0; k < 4; ++k) S += s[k] * expf(m[k] - M);

  const size_t o = ((size_t)b * ROWC + rc) * N + j;
  pm[o] = M;
  ps[o] = S;
}

// v[b,j] = M + log(S) combined over the ROWC partial (m,s) pairs.
__global__ __launch_bounds__(256) void colLSEcombine(const float* __restrict__ pm,
                                                     const float* __restrict__ ps,
                                                     float* __restrict__ v) {
  const int gid = blockIdx.x * 256 + threadIdx.x;     // 0 .. BATCH*N-1
  const int b = gid >> 10, j = gid & (N - 1);
  float M = -INFINITY;
  #pragma unroll
  for (int rc = 0; rc < ROWC; ++rc)
    M = fmaxf(M, pm[((size_t)b * ROWC + rc) * N + j]);
  float S = 0.f;
  #pragma unroll
  for (int rc = 0; rc < ROWC; ++rc) {
    const size_t o = ((size_t)b * ROWC + rc) * N + j;
    S += ps[o] * expf(pm[o] - M);
  }
  v[gid] = M + logf(S);
}

__global__ __launch_bounds__(256) void initV(float* __restrict__ v) {
  v[blockIdx.x * 256 + threadIdx.x] = 0.0f;
}

// out = exp(logits - u_i - v_j), float4 I/O.
__global__ __launch_bounds__(256) void finalizeExp(const float* __restrict__ logits,
                                                   const float* __restrict__ u,
                                                   const float* __restrict__ v,
                                                   float* __restrict__ out) {
  const size_t f = (size_t)blockIdx.x * 256 + threadIdx.x;   // float4 index
  const size_t e = f * 4;
  const int j4 = (int)(e & (size_t)(N - 1));
  const int i  = (int)((e >> 10) & (size_t)(N - 1));
  const int b  = (int)(e >> 20);
  float4 x  = *(const float4*)(logits + e);
  float4 vv = *(const float4*)(v + (size_t)b * N + j4);
  const float ui = u[(size_t)b * N + i];
  float4 o;
  o.x = expf(x.x - ui - vv.x);
  o.y = expf(x.y - ui - vv.y);
  o.z = expf(x.z - ui - vv.z);
  o.w = expf(x.w - ui - vv.w);
  *(float4*)(out + e) = o;
}

extern "C" void kernel_launch(void* const* d_in, const int* in_sizes, int n_in,
                              void* d_out, int out_size, void* d_ws, size_t ws_size,
                              hipStream_t stream) {
  const float* logits = (const float*)d_in[0];
  float* out = (float*)d_out;

  float* u  = (float*)d_ws;                          // BATCH*N
  float* v  = u  + (size_t)BATCH * N;                // BATCH*N
  float* pm = v  + (size_t)BATCH * N;                // BATCH*ROWC*N
  float* ps = pm + (size_t)BATCH * ROWC * N;         // BATCH*ROWC*N

  const dim3 blk(256);
  const dim3 gInit(BATCH * N / 256);
  const dim3 gU(N, BATCH);
  const dim3 gP(N / 256, ROWC, BATCH);
  const dim3 gC(BATCH * N / 256);
  const dim3 gF((unsigned)((size_t)BATCH * N * N / 4 / 256));

  initV<<<gInit, blk, 0, stream>>>(v);
  for (int it = 0; it < NUM_ITERS; ++it) {
    rowLSE<<<gU, blk, 0, stream>>>(logits, v, u);
    colLSEpart<<<gP, blk, 0, stream>>>(logits, u, pm, ps);
    colLSEcombine<<<gC, blk, 0, stream>>>(pm, ps, v);
  }
  finalizeExp<<<gF, blk, 0, stream>>>(logits, u, v, out);
}